// AttentionBlock_46093589021359
// MI455X (gfx1250) — compile-verified
//
#include <hip/hip_runtime.h>
#include <hip/hip_bf16.h>
#include <math.h>

// ---------------------------------------------------------------------------
// AttentionBlock for MI455X (gfx1250), pure fp32 via V_WMMA_F32_16X16X4_F32.
// B=8, C=512, H=W=32 -> L=1024, 32 groups, 8 heads, head_dim=64.
// ---------------------------------------------------------------------------

#define B_DIM 8
#define C_DIM 512
#define L_DIM 1024
#define NGROUPS 32
#define GSIZE 16            // channels per group = 512/32
#define NHEADS 8
#define HDIM 64
#define QKV_ROWS 1536       // 3*C
#define EPS 1e-5f

typedef __attribute__((ext_vector_type(2))) float v2f;
typedef __attribute__((ext_vector_type(8))) float v8f;

static __device__ __forceinline__ v8f wmma4(v2f a, v2f b, v8f c) {
    // D(16x16,f32) = A(16x4,f32) x B(4x16,f32) + C
    return __builtin_amdgcn_wmma_f32_16x16x4_f32(
        /*neg_a=*/false, a, /*neg_b=*/false, b,
        /*c_mod=*/(short)0, c, /*reuse_a=*/false, /*reuse_b=*/false);
}

// ---------------------------------------------------------------------------
// Kernel 1: GroupNorm. One block per (b, g). Group data is a contiguous
// 16*1024 float region. float4 passes + LDS tree reduction.
// ---------------------------------------------------------------------------
__global__ void __launch_bounds__(256, 1)
gn_kernel(const float* __restrict__ x,
          const float* __restrict__ gn_w,
          const float* __restrict__ gn_b,
          float* __restrict__ xn) {
    const int bg = blockIdx.x;            // 0..255
    const int b  = bg / NGROUPS;
    const int g  = bg % NGROUPS;
    const int tid = threadIdx.x;          // 0..255
    const size_t base = ((size_t)b * C_DIM + (size_t)g * GSIZE) * L_DIM;
    const int N  = GSIZE * L_DIM;         // 16384
    const int N4 = N >> 2;                // 4096 float4

    const float4* x4  = (const float4*)(x + base);
    float4*       xn4 = (float4*)(xn + base);

    float s = 0.f, ss = 0.f;
    for (int i = tid; i < N4; i += 256) {
        float4 v = x4[i];
        s  += v.x + v.y + v.z + v.w;
        ss += v.x * v.x + v.y * v.y + v.z * v.z + v.w * v.w;
    }
    __shared__ float red_s[256];
    __shared__ float red_q[256];
    red_s[tid] = s;
    red_q[tid] = ss;
    __syncthreads();
    for (int off = 128; off > 0; off >>= 1) {
        if (tid < off) {
            red_s[tid] += red_s[tid + off];
            red_q[tid] += red_q[tid + off];
        }
        __syncthreads();
    }
    const float mean = red_s[0] * (1.0f / N);
    const float var  = red_q[0] * (1.0f / N) - mean * mean;
    const float inv  = rsqrtf(var + EPS);

    for (int i = tid; i < N4; i += 256) {
        const int ch = g * GSIZE + (i >> 8);       // 256 float4 per channel
        const float sc = gn_w[ch] * inv;
        const float sh = gn_b[ch] - mean * sc;
        float4 v = x4[i];
        v.x = v.x * sc + sh;
        v.y = v.y * sc + sh;
        v.z = v.z * sc + sh;
        v.w = v.w * sc + sh;
        xn4[i] = v;
    }
}

// ---------------------------------------------------------------------------
// Kernel 2: GEMM  out[b, m, n] = sum_k W[m,k] * xn[b,k,n] (+ bias)
// One 16x16 tile per wave; 8 waves per 256-thread block cover 8 M-tiles.
// Double-buffered K-chunks of 16 (4 WMMA per chunk) to hide load latency.
// grid: (Mtiles/8, Ntiles=64, batch=8)
// ---------------------------------------------------------------------------
template <int MROWS, bool RESIDUAL>
__global__ void __launch_bounds__(256, 1)
gemm_kernel(const float* __restrict__ W,     // MROWS x 512
            const float* __restrict__ bias,  // MROWS
            const float* __restrict__ X,     // B x 512 x 1024
            const float* __restrict__ resid, // B x MROWS x 1024
            float* __restrict__ out) {       // B x MROWS x 1024
    const int lane   = threadIdx.x & 31;
    const int wave   = threadIdx.x >> 5;
    const int lane15 = lane & 15;
    const int khalf  = (lane < 16) ? 0 : 2;   // K sub-offset per A/B layout
    const int mhalf  = (lane < 16) ? 0 : 8;   // row half for C/D layout

    const int mt = blockIdx.x * 8 + wave;     // M tile index
    const int nt = blockIdx.y;                // N tile index
    const int b  = blockIdx.z;

    const int m0 = mt * 16;
    const int n0 = nt * 16;

    const float* Arow = W + (size_t)(m0 + lane15) * C_DIM;           // A row ptr
    const float* Bcol = X + (size_t)b * C_DIM * L_DIM + n0 + lane15; // B col ptr

    v2f abuf[2][4], bbuf[2][4];

    auto load_chunk = [&](int k, v2f* a4, v2f* b4) {
#pragma unroll
        for (int j = 0; j < 4; ++j) {
            const int ka = k + j * 4 + khalf;
            a4[j].x = Arow[ka];
            a4[j].y = Arow[ka + 1];
            b4[j].x = Bcol[(size_t)ka * L_DIM];
            b4[j].y = Bcol[(size_t)(ka + 1) * L_DIM];
        }
    };

    load_chunk(0, abuf[0], bbuf[0]);
    v8f acc = {};
    int cur = 0;
#pragma unroll 4
    for (int k = 0; k < C_DIM; k += 16) {
        if (k + 16 < C_DIM) {
            __builtin_prefetch(&Bcol[(size_t)(k + 32) * L_DIM], 0, 3);
            load_chunk(k + 16, abuf[cur ^ 1], bbuf[cur ^ 1]);
        }
#pragma unroll
        for (int j = 0; j < 4; ++j)
            acc = wmma4(abuf[cur][j], bbuf[cur][j], acc);
        cur ^= 1;
    }

    // Epilogue: D vgpr r holds (row = m0+r+mhalf, col = n0+lane15)
#pragma unroll
    for (int r = 0; r < 8; ++r) {
        const int row = m0 + r + mhalf;
        const size_t oidx = ((size_t)b * MROWS + row) * L_DIM + (n0 + lane15);
        float v = acc[r] + bias[row];
        if (RESIDUAL) v += resid[oidx];
        out[oidx] = v;
    }
}

// ---------------------------------------------------------------------------
// Kernel 3: flash attention per (b, head). qkv layout: (B, 1536, 1024) where
// head h of batch b owns rows [h*192, h*192+192): q | k | v, 64 rows each.
// Each wave owns a 16-row t-tile; block (8 waves) covers 128 t rows.
// grid: (L/128 = 8, B*NHEADS = 64)
// ---------------------------------------------------------------------------
__global__ void __launch_bounds__(256, 1)
attn_kernel(const float* __restrict__ qkv,
            float* __restrict__ aout) {     // B x 512 x 1024
    const int lane   = threadIdx.x & 31;
    const int wave   = threadIdx.x >> 5;
    const int lane15 = lane & 15;
    const int khalf  = (lane < 16) ? 0 : 2;
    const int mhalf  = (lane < 16) ? 0 : 8;

    const int bh = blockIdx.y;
    const int b  = bh >> 3;
    const int h  = bh & 7;
    const int t0 = blockIdx.x * 128 + wave * 16;

    const float* qptr = qkv + ((size_t)b * QKV_ROWS + h * 192) * L_DIM;
    const float* kptr = qptr + (size_t)HDIM * L_DIM;
    const float* vptr = qptr + (size_t)(2 * HDIM) * L_DIM;

    // Preload Q fragments (A[t][c] = q[c,t]); fold the 1/8 softmax scale in.
    v2f qf[16];
    {
        const int t = t0 + lane15;
#pragma unroll
        for (int c4 = 0; c4 < 16; ++c4) {
            const int k0 = c4 * 4 + khalf;
            qf[c4].x = qptr[(size_t)k0 * L_DIM + t] * 0.125f;
            qf[c4].y = qptr[(size_t)(k0 + 1) * L_DIM + t] * 0.125f;
        }
    }

    float mrow[8], lrow[8];
    v8f O[4];
#pragma unroll
    for (int r = 0; r < 8; ++r) { mrow[r] = -1e30f; lrow[r] = 0.f; }
#pragma unroll
    for (int ct = 0; ct < 4; ++ct) O[ct] = (v8f){};

    __shared__ float plds[8][16][17];   // per-wave 16x16 P tile, padded

    for (int s0 = 0; s0 < L_DIM; s0 += 16) {
        // ---- batch-load all 16 K B-fragments, then the WMMA chain.
        v2f bf[16];
#pragma unroll
        for (int c4 = 0; c4 < 16; ++c4) {
            const int k0 = c4 * 4 + khalf;
            bf[c4].x = kptr[(size_t)k0 * L_DIM + s0 + lane15];
            bf[c4].y = kptr[(size_t)(k0 + 1) * L_DIM + s0 + lane15];
        }
        v8f S = {};
#pragma unroll
        for (int c4 = 0; c4 < 16; ++c4)
            S = wmma4(qf[c4], bf[c4], S);

        // ---- preload V fragments; latency hides under the softmax VALU.
        // vf[ct*4+j]: B[s][c] = v[c, s]
        v2f vf[16];
#pragma unroll
        for (int ct = 0; ct < 4; ++ct) {
            const int c = ct * 16 + lane15;
#pragma unroll
            for (int j = 0; j < 4; ++j) {
                const int sk = s0 + j * 4 + khalf;
                vf[ct * 4 + j].x = vptr[(size_t)c * L_DIM + sk];
                vf[ct * 4 + j].y = vptr[(size_t)c * L_DIM + sk + 1];
            }
        }

        // ---- online softmax, in place (S -> P); O rescale fused per row.
#pragma unroll
        for (int r = 0; r < 8; ++r) {
            const float sv = S[r];
            float mx = sv;
            mx = fmaxf(mx, __shfl_xor(mx, 1, 16));
            mx = fmaxf(mx, __shfl_xor(mx, 2, 16));
            mx = fmaxf(mx, __shfl_xor(mx, 4, 16));
            mx = fmaxf(mx, __shfl_xor(mx, 8, 16));
            const float mnew = fmaxf(mrow[r], mx);
            const float corr = __expf(mrow[r] - mnew);
            const float p = __expf(sv - mnew);
            float rs = p;
            rs += __shfl_xor(rs, 1, 16);
            rs += __shfl_xor(rs, 2, 16);
            rs += __shfl_xor(rs, 4, 16);
            rs += __shfl_xor(rs, 8, 16);
            lrow[r] = lrow[r] * corr + rs;
            mrow[r] = mnew;
            S[r]    = p;              // S now holds P
            O[0][r] *= corr;
            O[1][r] *= corr;
            O[2][r] *= corr;
            O[3][r] *= corr;
        }

        // ---- layout convert P (C/D layout -> A layout) through LDS
#pragma unroll
        for (int r = 0; r < 8; ++r)
            plds[wave][r + mhalf][lane15] = S[r];
        __syncthreads();

        // ---- O += P x V^T. A-frag depends only on j: load once, reuse for
        // all 4 c-tiles; 4 independent accumulators issue back-to-back.
#pragma unroll
        for (int j = 0; j < 4; ++j) {
            v2f af;
            af.x = plds[wave][lane15][j * 4 + khalf];
            af.y = plds[wave][lane15][j * 4 + khalf + 1];
#pragma unroll
            for (int ct = 0; ct < 4; ++ct)
                O[ct] = wmma4(af, vf[ct * 4 + j], O[ct]);
        }
        __syncthreads();
    }

    // ---- normalize and store: a[b, h*64 + c, t]
#pragma unroll
    for (int ct = 0; ct < 4; ++ct) {
        const int c = ct * 16 + lane15;
        const float* Oc = (const float*)&O[ct];
#pragma unroll
        for (int r = 0; r < 8; ++r) {
            const int t = t0 + r + mhalf;
            aout[((size_t)b * C_DIM + h * HDIM + c) * L_DIM + t] = Oc[r] / lrow[r];
        }
    }
}

// ---------------------------------------------------------------------------
// Host launcher
// ---------------------------------------------------------------------------
extern "C" void kernel_launch(void* const* d_in, const int* in_sizes, int n_in,
                              void* d_out, int out_size, void* d_ws, size_t ws_size,
                              hipStream_t stream) {
    const float* x      = (const float*)d_in[0];
    const float* gn_w   = (const float*)d_in[1];
    const float* gn_b   = (const float*)d_in[2];
    const float* qkv_w  = (const float*)d_in[3];
    const float* qkv_b  = (const float*)d_in[4];
    const float* proj_w = (const float*)d_in[5];
    const float* proj_b = (const float*)d_in[6];
    float* out = (float*)d_out;

    char* ws = (char*)d_ws;
    float* xn   = (float*)(ws);                                   // 16 MB
    float* qkv  = (float*)(ws + (size_t)16 * 1024 * 1024);        // 48 MB
    float* attn = (float*)(ws + (size_t)64 * 1024 * 1024);        // 16 MB

    // 1. GroupNorm
    gn_kernel<<<dim3(B_DIM * NGROUPS), dim3(256), 0, stream>>>(x, gn_w, gn_b, xn);

    // 2. QKV GEMM: M=1536 (96 tiles / 8 per block = 12), N tiles = 64, batch 8
    gemm_kernel<QKV_ROWS, false><<<dim3(12, 64, 8), dim3(256), 0, stream>>>(
        qkv_w, qkv_b, xn, nullptr, qkv);

    // 3. Attention: 8 t-blocks of 128, 64 (batch, head) pairs
    attn_kernel<<<dim3(8, 64), dim3(256), 0, stream>>>(qkv, attn);

    // 4. Proj GEMM + bias + residual: M=512 (32 tiles / 8 = 4)
    gemm_kernel<C_DIM, true><<<dim3(4, 64, 8), dim3(256), 0, stream>>>(
        proj_w, proj_b, attn, x, out);
}